// SpConvUNet_32787780337855
// MI455X (gfx1250) — compile-verified
//
#include <hip/hip_runtime.h>

// ---------------------------------------------------------------------------
// SpConvUNet on MI455X (gfx1250): every conv is an implicit GEMM on the
// V_WMMA_F32_16X16X4_F32 tensor-pipe path (full f32 precision, f32 accum).
// One wave32 computes a 64-voxel x 32-cout tile (8 accumulators): 8 WMMAs per
// 6 b64 loads per K-step. Weights are pre-packed (channel-pair interleave) so
// B fragments are single b64 loads. Spatial dims are powers of two, so all
// voxel decompositions are mask/shift. Conv kernels are templated on
// <KS,STRIDE,PAD>; pad-0 variants have no bounds checks. BN stats use a
// deterministic two-stage reduction. Concat via channel-strided stores;
// residual add is a conv accumulate epilogue.
// ---------------------------------------------------------------------------

typedef __attribute__((ext_vector_type(2))) float v2f;
typedef __attribute__((ext_vector_type(8))) float v8f;

#define BN_EPS 1e-4f
#define NSLAB 32

__device__ __forceinline__ v8f wmma_f32_k4(v2f a, v2f b, v8f c) {
  // 8-arg pattern: (neg_a, A, neg_b, B, c_mod, C, reuse_a, reuse_b)
  return __builtin_amdgcn_wmma_f32_16x16x4_f32(
      false, a, false, b, (short)0, c, false, false);
}

// ---------------- BN stats: per-channel sum / sumsq ------------------------
__global__ __launch_bounds__(256) void k_bn_partial(
    const float* __restrict__ x, float* __restrict__ part, int nvox, int C) {
  int c = blockIdx.x;
  int slab = blockIdx.y;
  int nslab = gridDim.y;
  float s = 0.f, s2 = 0.f;
  for (int v = slab * blockDim.x + threadIdx.x; v < nvox;
       v += nslab * blockDim.x) {
    float t = x[(size_t)v * C + c];
    s += t;
    s2 += t * t;
  }
  for (int o = 16; o > 0; o >>= 1) {
    s += __shfl_down(s, o, 32);
    s2 += __shfl_down(s2, o, 32);
  }
  __shared__ float red0[8];
  __shared__ float red1[8];
  int lane = threadIdx.x & 31, wv = threadIdx.x >> 5;
  if (lane == 0) { red0[wv] = s; red1[wv] = s2; }
  __syncthreads();
  if (threadIdx.x == 0) {
    float a = 0.f, b = 0.f;
    int nw = blockDim.x >> 5;
    for (int i = 0; i < nw; ++i) { a += red0[i]; b += red1[i]; }
    part[(size_t)(2 * c) * nslab + slab] = a;
    part[(size_t)(2 * c + 1) * nslab + slab] = b;
  }
}

__global__ __launch_bounds__(64) void k_bn_finalize(
    const float* __restrict__ part, float* __restrict__ stats, int C,
    int nslab) {
  int c = blockIdx.x * blockDim.x + threadIdx.x;
  if (c < C) {
    float s = 0.f, s2 = 0.f;
    for (int i = 0; i < nslab; ++i) {
      s += part[(size_t)(2 * c) * nslab + i];
      s2 += part[(size_t)(2 * c + 1) * nslab + i];
    }
    stats[2 * c] = s;
    stats[2 * c + 1] = s2;
  }
}

__global__ __launch_bounds__(256) void k_bn_apply(
    const float* __restrict__ x, const float* __restrict__ stats,
    float* __restrict__ y, size_t n, int C, float invN) {
  size_t i = (size_t)blockIdx.x * blockDim.x + threadIdx.x;
  if (i < n) {
    int c = (int)(i % C);
    float mean = stats[2 * c] * invN;
    float var = stats[2 * c + 1] * invN - mean * mean;
    float sc = rsqrtf(var + BN_EPS);
    float v = (x[i] - mean) * sc;
    y[i] = v > 0.f ? v : 0.f;
  }
}

// ---------------- channel-strided copy (plain copy / concat half) ----------
__global__ __launch_bounds__(256) void k_copy_chan(
    const float* __restrict__ src, float* __restrict__ dst, size_t nvox, int C,
    int ldd, int coff) {
  size_t i = (size_t)blockIdx.x * blockDim.x + threadIdx.x;
  if (i < nvox * (size_t)C) {
    size_t v = i / C;
    int c = (int)(i % C);
    dst[v * ldd + coff + c] = src[i];
  }
}

// ---------------- weight packing: channel-pair interleave ------------------
// src (tap, ci, co) row-major -> pk[((tap*Cin/2 + p)*Cout + co)] = float2
// { src[tap, 2p, co], src[tap, 2p+1, co] }  => B frag becomes one b64 load.
__global__ __launch_bounds__(256) void k_pack_w(
    const float* __restrict__ w, float* __restrict__ pk, int pairs, int Cin,
    int Cout) {
  int i = blockIdx.x * blockDim.x + threadIdx.x;
  if (i < pairs) {
    int co = i % Cout;
    int row = i / Cout;
    int half = Cin >> 1;
    int p = row % half;
    int tap = row / half;
    const float* src = w + ((size_t)(tap * Cin + 2 * p)) * Cout + co;
    pk[2 * (size_t)i] = src[0];
    pk[2 * (size_t)i + 1] = src[Cout];
  }
}

// ---------------- generic conv (KS^3, STRIDE, PAD) via WMMA f32 ------------
// grid.x = out_voxels/64 (4 M tiles), grid.y = Cout/32 (2 N tiles), block=32.
// A frag (16x4 f32): lanes0-15 hold K=k0,k0+1; lanes16-31 hold K=k0+2,k0+3.
// B frag (4x16) mirrored. C/D: 8 VGPRs, lanes16-31 hold rows M=r+8.
template <int KS, int STRIDE, int PAD>
__global__ __launch_bounds__(32) void k_conv_wmma(
    const float* __restrict__ x, const float* __restrict__ pk,
    float* __restrict__ y, int D, int H, int W, int OHsh, int OWsh, int Cin,
    int Cout, int ldy, int coff, int addF) {
  const int lane = threadIdx.x;
  const int mT = blockIdx.x, nT = blockIdx.y;
  const int n = lane & 15;
  const int hi = lane >> 4;    // 0 | 1
  const int khalf = hi << 1;   // 0 | 2
  const int co = nT * 32 + n;  // first N-tile column
  const int OW = 1 << OWsh, OH = 1 << OHsh;

  int ow[4], oh[4], od[4];
#pragma unroll
  for (int j = 0; j < 4; ++j) {
    int m = mT * 64 + 16 * j + n;
    ow[j] = m & (OW - 1);
    int t = m >> OWsh;
    oh[j] = t & (OH - 1);
    od[j] = t >> OHsh;
  }

  v8f acc[4][2];
#pragma unroll
  for (int j = 0; j < 4; ++j)
#pragma unroll
    for (int q = 0; q < 2; ++q) acc[j][q] = (v8f){0.f, 0.f, 0.f, 0.f,
                                                  0.f, 0.f, 0.f, 0.f};

#pragma unroll
  for (int kd = 0; kd < KS; ++kd) {
#pragma unroll
    for (int kh = 0; kh < KS; ++kh) {
#pragma unroll
      for (int kw = 0; kw < KS; ++kw) {
        const int tap = (kd * KS + kh) * KS + kw;
        const float* xa[4];
        bool inb[4];
#pragma unroll
        for (int j = 0; j < 4; ++j) {
          int zd = od[j] * STRIDE + kd - PAD;
          int zh = oh[j] * STRIDE + kh - PAD;
          int zw = ow[j] * STRIDE + kw - PAD;
          inb[j] = true;
          if (PAD != 0) {
            inb[j] = ((unsigned)zd < (unsigned)D) &
                     ((unsigned)zh < (unsigned)H) &
                     ((unsigned)zw < (unsigned)W);
          }
          xa[j] = x + ((size_t)((zd * H) + zh) * W + zw) * Cin;
        }
        const float* pw = pk + (size_t)tap * (Cin >> 1) * Cout * 2;
#pragma unroll 2
        for (int k0 = 0; k0 < Cin; k0 += 4) {
          int kk = k0 + khalf;
          size_t q = (size_t)((k0 >> 1) + hi) * Cout + co;
          v2f b0 = *(const v2f*)(pw + 2 * q);
          v2f b1 = *(const v2f*)(pw + 2 * q + 32);
          v2f a[4];
#pragma unroll
          for (int j = 0; j < 4; ++j)
            a[j] = inb[j] ? *(const v2f*)(xa[j] + kk) : (v2f){0.f, 0.f};
#pragma unroll
          for (int j = 0; j < 4; ++j) {
            acc[j][0] = wmma_f32_k4(a[j], b0, acc[j][0]);
            acc[j][1] = wmma_f32_k4(a[j], b1, acc[j][1]);
          }
        }
      }
    }
  }

#pragma unroll
  for (int j = 0; j < 4; ++j) {
    int mb = mT * 64 + 16 * j + (hi << 3);
#pragma unroll
    for (int r = 0; r < 8; ++r) {
      size_t o = (size_t)(mb + r) * ldy + coff + co;
      if (addF) {
        y[o] += acc[j][0][r];
        y[o + 16] += acc[j][1][r];
      } else {
        y[o] = acc[j][0][r];
        y[o + 16] = acc[j][1][r];
      }
    }
  }
}

// ---------------- k=2 s=2 transposed conv: 8 parity taps, 1x1 GEMM each ----
// grid.x = in_voxels/64, grid.y = Cout/32, grid.z = 8 (tap = kd*4+kh*2+kw).
__global__ __launch_bounds__(32) void k_convT_wmma(
    const float* __restrict__ x, const float* __restrict__ pk,
    float* __restrict__ y, int Hsh, int Wsh, int Cin, int Cout, int ldy,
    int coff) {
  const int lane = threadIdx.x;
  const int mT = blockIdx.x, nT = blockIdx.y, tap = blockIdx.z;
  const int n = lane & 15;
  const int hi = lane >> 4;
  const int khalf = hi << 1;
  const int co = nT * 32 + n;
  const int kw = tap & 1, kh = (tap >> 1) & 1, kd = (tap >> 2) & 1;
  const int W = 1 << Wsh, H = 1 << Hsh;

  const float* xa0 = x + (size_t)(mT * 64 + n) * Cin;
  const float* pw = pk + (size_t)tap * (Cin >> 1) * Cout * 2;

  v8f acc[4][2];
#pragma unroll
  for (int j = 0; j < 4; ++j)
#pragma unroll
    for (int q = 0; q < 2; ++q) acc[j][q] = (v8f){0.f, 0.f, 0.f, 0.f,
                                                  0.f, 0.f, 0.f, 0.f};
#pragma unroll 2
  for (int k0 = 0; k0 < Cin; k0 += 4) {
    int kk = k0 + khalf;
    size_t q = (size_t)((k0 >> 1) + hi) * Cout + co;
    v2f b0 = *(const v2f*)(pw + 2 * q);
    v2f b1 = *(const v2f*)(pw + 2 * q + 32);
#pragma unroll
    for (int j = 0; j < 4; ++j) {
      v2f a = *(const v2f*)(xa0 + (size_t)16 * j * Cin + kk);
      acc[j][0] = wmma_f32_k4(a, b0, acc[j][0]);
      acc[j][1] = wmma_f32_k4(a, b1, acc[j][1]);
    }
  }

#pragma unroll
  for (int j = 0; j < 4; ++j) {
    int mb = mT * 64 + 16 * j + (hi << 3);
#pragma unroll
    for (int r = 0; r < 8; ++r) {
      int mm = mb + r;
      int w2 = mm & (W - 1);
      int t2 = mm >> Wsh;
      int h2 = t2 & (H - 1);
      int d2 = t2 >> Hsh;
      size_t ovox =
          ((size_t)(2 * d2 + kd) * (2 * H) + (2 * h2 + kh)) * (2 * W) +
          (2 * w2 + kw);
      y[ovox * ldy + coff + co] = acc[j][0][r];
      y[ovox * ldy + coff + co + 16] = acc[j][1][r];
    }
  }
}

// ============================ host orchestration ===========================

static inline int ilog2(int v) { return __builtin_ctz((unsigned)v); }

static void launch_bnrelu(const float* x, float* y, float* part, float* stats,
                          int nvox, int C, hipStream_t s) {
  dim3 g(C, NSLAB);
  k_bn_partial<<<g, 256, 0, s>>>(x, part, nvox, C);
  k_bn_finalize<<<(C + 63) / 64, 64, 0, s>>>(part, stats, C, NSLAB);
  size_t n = (size_t)nvox * C;
  k_bn_apply<<<(unsigned)((n + 255) / 256), 256, 0, s>>>(x, stats, y, n, C,
                                                         1.0f / (float)nvox);
}

static void pack_weights(const float* w, float* pk, int taps, int Cin, int Cout,
                         hipStream_t s) {
  int pairs = taps * (Cin >> 1) * Cout;
  k_pack_w<<<(pairs + 255) / 256, 256, 0, s>>>(w, pk, pairs, Cin, Cout);
}

static void launch_conv(const float* x, const float* w, float* pk, float* y,
                        int D, int Cin, int Cout, int ks, int OD, int ldy,
                        int coff, int addF, hipStream_t s) {
  pack_weights(w, pk, ks * ks * ks, Cin, Cout, s);
  dim3 g((unsigned)((OD * OD * OD) / 64), (unsigned)(Cout / 32));
  int osh = ilog2(OD);
  if (ks == 3)
    k_conv_wmma<3, 1, 1><<<g, 32, 0, s>>>(x, pk, y, D, D, D, osh, osh, Cin,
                                          Cout, ldy, coff, addF);
  else if (ks == 2)
    k_conv_wmma<2, 2, 0><<<g, 32, 0, s>>>(x, pk, y, D, D, D, osh, osh, Cin,
                                          Cout, ldy, coff, addF);
  else
    k_conv_wmma<1, 1, 0><<<g, 32, 0, s>>>(x, pk, y, D, D, D, osh, osh, Cin,
                                          Cout, ldy, coff, addF);
}

static void launch_convT(const float* x, const float* w, float* pk, float* y,
                         int D, int Cin, int Cout, int ldy, int coff,
                         hipStream_t s) {
  pack_weights(w, pk, 8, Cin, Cout, s);
  dim3 g((unsigned)((D * D * D) / 64), (unsigned)(Cout / 32), 8);
  int sh = ilog2(D);
  k_convT_wmma<<<g, 32, 0, s>>>(x, pk, y, sh, sh, Cin, Cout, ldy, coff);
}

static void launch_copy(const float* src, float* dst, size_t nvox, int C,
                        int ldd, int coff, hipStream_t s) {
  size_t n = nvox * (size_t)C;
  k_copy_chan<<<(unsigned)((n + 255) / 256), 256, 0, s>>>(src, dst, nvox, C,
                                                          ldd, coff);
}

// out = conv2(bnrelu(conv1(bnrelu(X)))) + idn(X); idn = copy or 1x1 conv (wi)
static void resblock(const float* X, float* Y, float* T, float* U, float* pk,
                     float* part, float* stats, const float* w1,
                     const float* w2, const float* wi, int D, int Cin,
                     int Cout, hipStream_t s) {
  int nv = D * D * D;
  if (wi)
    launch_conv(X, wi, pk, Y, D, Cin, Cout, 1, D, Cout, 0, 0, s);
  else
    launch_copy(X, Y, (size_t)nv, Cout, Cout, 0, s);
  launch_bnrelu(X, T, part, stats, nv, Cin, s);
  launch_conv(T, w1, pk, U, D, Cin, Cout, 3, D, Cout, 0, 0, s);
  launch_bnrelu(U, T, part, stats, nv, Cout, s);  // T >= nv*Cout, reuse
  launch_conv(T, w2, pk, Y, D, Cout, Cout, 3, D, Cout, 0, 1, s);
}

extern "C" void kernel_launch(void* const* d_in, const int* in_sizes, int n_in,
                              void* d_out, int out_size, void* d_ws,
                              size_t ws_size, hipStream_t stream) {
  (void)in_sizes; (void)n_in; (void)out_size; (void)ws_size;
  const int C[5] = {32, 64, 96, 128, 160};
  const float* xin = (const float*)d_in[0];
  // flattened param order (dict insertion): per level l<4:
  //  h0.w1,h0.w2,h1.w1,h1.w2, down, up, t0.w1,t0.w2,t0.wi, t1.w1,t1.w2 (11)
  // level4: h0.w1,h0.w2,h1.w1,h1.w2 (4). base[l] = 1 + 11*l.
  auto Wp = [&](int i) { return (const float*)d_in[i]; };

  float* ws = (float*)d_ws;
  size_t off = 0;
  auto alloc = [&](size_t n) {
    float* p = ws + off;
    off += n;
    return p;
  };
  float* part = alloc(2 * 256 * NSLAB);
  float* stats = alloc(2 * 256);
  float* pk = alloc(27 * 256 * 128);  // packed-weight scratch (max conv)
  float *A[5], *B[5], *T1[5], *T2[5], *SK[4], *CC[4], *TC[4], *Z[5];
  for (int l = 0; l < 5; ++l) {
    int Dd = 64 >> l;
    size_t V = (size_t)Dd * Dd * Dd;
    A[l] = alloc(V * C[l]);
    B[l] = alloc(V * C[l]);
    T1[l] = alloc(V * C[l]);
    T2[l] = alloc(V * C[l]);
    if (l < 4) {
      SK[l] = alloc(V * C[l]);
      CC[l] = alloc(V * 2 * C[l]);
      TC[l] = alloc(V * 2 * C[l]);
    }
    if (l > 0) Z[l] = alloc(V * C[l]);
  }

  // -------- down path: heads + stride-2 down conv --------
  for (int l = 0; l < 4; ++l) {
    int Dd = 64 >> l, c = C[l], cn = C[l + 1];
    int bs = 1 + 11 * l;
    const float* src = (l == 0) ? xin : A[l];
    resblock(src, B[l], T1[l], T2[l], pk, part, stats, Wp(bs + 0), Wp(bs + 1),
             nullptr, Dd, c, c, stream);
    resblock(B[l], SK[l], T1[l], T2[l], pk, part, stats, Wp(bs + 2),
             Wp(bs + 3), nullptr, Dd, c, c, stream);
    int nv = Dd * Dd * Dd, Dn = Dd / 2;
    launch_bnrelu(SK[l], T1[l], part, stats, nv, c, stream);
    launch_conv(T1[l], Wp(bs + 4), pk, A[l + 1], Dd, c, cn, 2, Dn, cn, 0, 0,
                stream);
  }

  // -------- bottom level 4 (heads only) --------
  {
    int bs = 1 + 11 * 4;
    resblock(A[4], B[4], T1[4], T2[4], pk, part, stats, Wp(bs + 0), Wp(bs + 1),
             nullptr, 4, C[4], C[4], stream);
    resblock(B[4], Z[4], T1[4], T2[4], pk, part, stats, Wp(bs + 2), Wp(bs + 3),
             nullptr, 4, C[4], C[4], stream);
  }

  // -------- up path: bnrelu -> convT (into concat hi half) -> tails --------
  for (int l = 3; l >= 0; --l) {
    int Dd = 64 >> l, c = C[l], cn = C[l + 1], Dn = Dd / 2;
    int bs = 1 + 11 * l;
    int Vn = Dn * Dn * Dn;
    size_t V = (size_t)Dd * Dd * Dd;
    launch_bnrelu(Z[l + 1], T1[l + 1], part, stats, Vn, cn, stream);
    launch_convT(T1[l + 1], Wp(bs + 5), pk, CC[l], Dn, cn, c, 2 * c, c,
                 stream);
    launch_copy(SK[l], CC[l], V, c, 2 * c, 0, stream);  // concat low half
    float* Zdst = (l == 0) ? (float*)d_out : Z[l];
    resblock(CC[l], B[l], TC[l], T2[l], pk, part, stats, Wp(bs + 6),
             Wp(bs + 7), Wp(bs + 8), Dd, 2 * c, c, stream);
    resblock(B[l], Zdst, T1[l], T2[l], pk, part, stats, Wp(bs + 9),
             Wp(bs + 10), nullptr, Dd, c, c, stream);
  }
}